// AtomFeaturizer_27616639713567
// MI455X (gfx1250) — compile-verified
//
#include <hip/hip_runtime.h>
#include <hip/hip_bf16.h>
#include <stdint.h>

// ---------------------------------------------------------------------------
// AtomFeaturizer for MI455X (gfx1250, wave32)
//   out[a, :] = sum_t table_t[idx_t[a], :] + scalars[a,:] @ W^T + b
// Store-bandwidth-bound (~550MB -> ~23us floor @ 23.3 TB/s).
//   - tables (38.4 KB) DMA'd to LDS once per block via tensor_load_to_lds (TDM)
//   - linear(3->128)+bias done with V_WMMA_F32_16X16X4_F32 (K=3 lane carries 1.0
//     so the bias rides in B's K=3 row), C = embedding sum gathered from LDS
//   - non-temporal streaming loads/stores for the 1M-atom vectors
// ---------------------------------------------------------------------------

typedef float v2f __attribute__((ext_vector_type(2)));
typedef float v8f __attribute__((ext_vector_type(8)));
typedef unsigned int v4u __attribute__((ext_vector_type(4)));
typedef int v4i __attribute__((ext_vector_type(4)));
typedef int v8i __attribute__((ext_vector_type(8)));

#define D 128
#define ROWS0 46
#define ROWS1 6
#define ROWS2 11
#define ROWS3 3
#define ROWS4 5
#define ROWS5 4
#define OFF0 0
#define OFF1 (OFF0 + ROWS0 * D) /* 5888 */
#define OFF2 (OFF1 + ROWS1 * D) /* 6656 */
#define OFF3 (OFF2 + ROWS2 * D) /* 8064 */
#define OFF4 (OFF3 + ROWS3 * D) /* 8448 */
#define OFF5 (OFF4 + ROWS4 * D) /* 9088 */
#define LDS_FLOATS (OFF5 + ROWS5 * D) /* 9600 floats = 38400 B */

#if __has_builtin(__builtin_amdgcn_tensor_load_to_lds)
#define HAVE_TDM 1
#else
#define HAVE_TDM 0
#endif

#if __has_builtin(__builtin_amdgcn_wmma_f32_16x16x4_f32)
#define HAVE_WMMA 1
#else
#define HAVE_WMMA 0
#endif

#if HAVE_TDM
// Issue one TDM descriptor copying a [rows x 128] f32 table into LDS.
__device__ __forceinline__ void tdm_load_table(const float* gsrc,
                                               uint32_t lds_byte_addr,
                                               int rows) {
  uint64_t ga = (uint64_t)(uintptr_t)gsrc;
  v4u g0;
  g0.x = 1u;                                   // count=1, is_restore=0, gather off
  g0.y = lds_byte_addr;                        // lds_addr (bytes)
  g0.z = (uint32_t)(ga & 0xFFFFFFFFu);         // global_addr[31:0]
  g0.w = (uint32_t)((ga >> 32) & 0x1FFFFFFu)   // global_addr[56:32]
         | (2u << 30);                         // type = 2 ("image")
  v8i g1;
  g1[0] = (2 << 16);                  // wg_mask=0, data_size=2 (4B)
  g1[1] = (int)(128u << 16);          // tensor_dim0 = 128
  g1[2] = (int)((uint32_t)rows << 16);// tensor_dim1 = rows
  g1[3] = (int)(128u << 16);          // tile_dim0 = 128
  g1[4] = rows;                       // tile_dim1 = rows, tile_dim2 = 0
  g1[5] = 128;                        // tensor_dim0_stride = 128
  g1[6] = 0;
  g1[7] = 0;
  v4i z4 = {0, 0, 0, 0};
#if __clang_major__ >= 23
  v8i z8 = {0, 0, 0, 0, 0, 0, 0, 0};
  __builtin_amdgcn_tensor_load_to_lds(g0, g1, z4, z4, z8, 0);
#else
  __builtin_amdgcn_tensor_load_to_lds(g0, g1, z4, z4, 0);
#endif
}
#endif

__global__ __launch_bounds__(256) void atom_feat_kernel(
    const int* __restrict__ i_atom, const int* __restrict__ i_deg,
    const int* __restrict__ i_chg, const int* __restrict__ i_hyb,
    const int* __restrict__ i_nh, const int* __restrict__ i_chi,
    const float* __restrict__ scalars, const float* __restrict__ T0,
    const float* __restrict__ T1, const float* __restrict__ T2,
    const float* __restrict__ T3, const float* __restrict__ T4,
    const float* __restrict__ T5, const float* __restrict__ W,
    const float* __restrict__ bias, float* __restrict__ out, int n) {
  __shared__ float ldsT[LDS_FLOATS];

  // ---- Phase 0: stage all 6 embedding tables into LDS -------------------
#if HAVE_TDM
  if ((threadIdx.x >> 5) == 0) {
    uint32_t lbase = (uint32_t)(uintptr_t)&ldsT[0];
    tdm_load_table(T0, lbase + OFF0 * 4, ROWS0);
    tdm_load_table(T1, lbase + OFF1 * 4, ROWS1);
    tdm_load_table(T2, lbase + OFF2 * 4, ROWS2);
    tdm_load_table(T3, lbase + OFF3 * 4, ROWS3);
    tdm_load_table(T4, lbase + OFF4 * 4, ROWS4);
    tdm_load_table(T5, lbase + OFF5 * 4, ROWS5);
    __builtin_amdgcn_s_wait_tensorcnt(0);
  }
#else
  {
    const float* tabs[6] = {T0, T1, T2, T3, T4, T5};
    const int offs[6] = {OFF0, OFF1, OFF2, OFF3, OFF4, OFF5};
    const int cnts[6] = {ROWS0 * D, ROWS1 * D, ROWS2 * D,
                         ROWS3 * D, ROWS4 * D, ROWS5 * D};
    for (int t = 0; t < 6; ++t)
      for (int j = (int)threadIdx.x; j < cnts[t]; j += (int)blockDim.x)
        ldsT[offs[t] + j] = tabs[t][j];
  }
#endif
  __syncthreads();

  const int lane = (int)(threadIdx.x & 31);
  const int laneLo = lane & 15;  // WMMA N index / atom-within-tile for loads
  const int laneHi = lane >> 4;  // 0: rows 0..7 | K 0,1 ; 1: rows 8..15 | K 2,3

  // ---- Hoisted B operand: [4 x 16] per 16-col block ---------------------
  // K=0..2 rows hold W^T, K=3 row holds the bias (A's K=3 element is 1.0).
#if HAVE_WMMA
  v2f Bv[8];
#pragma unroll
  for (int nb = 0; nb < 8; ++nb) {
    int col = nb * 16 + laneLo;
    float w0 = W[col * 3 + 0], w1 = W[col * 3 + 1];
    float w2 = W[col * 3 + 2], bb = bias[col];
    Bv[nb].x = laneHi ? w2 : w0;  // K = 2*laneHi + 0
    Bv[nb].y = laneHi ? bb : w1;  // K = 2*laneHi + 1
  }
#else
  float wc[8][3], bc[8];
#pragma unroll
  for (int nb = 0; nb < 8; ++nb) {
    int col = nb * 16 + laneLo;
    wc[nb][0] = W[col * 3 + 0];
    wc[nb][1] = W[col * 3 + 1];
    wc[nb][2] = W[col * 3 + 2];
    bc[nb] = bias[col];
  }
#endif

  const int nTiles = n >> 4;  // full 16-atom tiles
  const int wavesPerBlk = (int)(blockDim.x >> 5);
  const int gw = (int)blockIdx.x * wavesPerBlk + (int)(threadIdx.x >> 5);
  const int wstride = (int)gridDim.x * wavesPerBlk;

  for (int tile = gw; tile < nTiles; tile += wstride) {
    const int base = tile << 4;
    const int atomA = base + laneLo;

    // A operand: lane supplies K = 2*laneHi + {0,1} of row M = laneLo.
    float s0 = __builtin_nontemporal_load(scalars + atomA * 3 + 0);
    float s1 = __builtin_nontemporal_load(scalars + atomA * 3 + 1);
    float s2 = __builtin_nontemporal_load(scalars + atomA * 3 + 2);

    // Indices for this lane's atom; redistributed across rows via shfl.
    int j0 = __builtin_nontemporal_load(i_atom + atomA);
    int j1 = __builtin_nontemporal_load(i_deg + atomA);
    int j2 = __builtin_nontemporal_load(i_chg + atomA);
    int j3 = __builtin_nontemporal_load(i_hyb + atomA);
    int j4 = __builtin_nontemporal_load(i_nh + atomA);
    int j5 = __builtin_nontemporal_load(i_chi + atomA);

    // ---- C = embedding sum, gathered from LDS in WMMA C-layout ----------
    float c[8][8];
#pragma unroll
    for (int v = 0; v < 8; ++v) {
      int m = v + 8 * laneHi;  // output row this lane owns in VGPR v
      int r0 = OFF0 + __shfl(j0, m, 32) * D;
      int r1 = OFF1 + __shfl(j1, m, 32) * D;
      int r2 = OFF2 + __shfl(j2, m, 32) * D;
      int r3 = OFF3 + __shfl(j3, m, 32) * D;
      int r4 = OFF4 + __shfl(j4, m, 32) * D;
      int r5 = OFF5 + __shfl(j5, m, 32) * D;
#pragma unroll
      for (int nb = 0; nb < 8; ++nb) {
        int col = nb * 16 + laneLo;
        c[nb][v] = ldsT[r0 + col] + ldsT[r1 + col] + ldsT[r2 + col] +
                   ldsT[r3 + col] + ldsT[r4 + col] + ldsT[r5 + col];
      }
    }

#if HAVE_WMMA
    v2f a;
    a.x = laneHi ? s2 : s0;
    a.y = laneHi ? 1.0f : s1;  // K=3 multiplies the bias row of B
#pragma unroll
    for (int nb = 0; nb < 8; ++nb) {
      v8f cv = {c[nb][0], c[nb][1], c[nb][2], c[nb][3],
                c[nb][4], c[nb][5], c[nb][6], c[nb][7]};
      v8f d = __builtin_amdgcn_wmma_f32_16x16x4_f32(
          false, a, false, Bv[nb], (short)0, cv, false, false);
#pragma unroll
      for (int v = 0; v < 8; ++v) {
        int row = base + v + 8 * laneHi;
        __builtin_nontemporal_store(d[v], out + row * D + nb * 16 + laneLo);
      }
    }
#else
    // VALU fallback: shuffle scalars to the row this lane owns.
#pragma unroll
    for (int v = 0; v < 8; ++v) {
      int m = v + 8 * laneHi;
      float sm0 = __shfl(s0, m, 32);
      float sm1 = __shfl(s1, m, 32);
      float sm2 = __shfl(s2, m, 32);
#pragma unroll
      for (int nb = 0; nb < 8; ++nb) {
        float d = c[nb][v] + sm0 * wc[nb][0] + sm1 * wc[nb][1] +
                  sm2 * wc[nb][2] + bc[nb];
        int row = base + v + 8 * laneHi;
        __builtin_nontemporal_store(d, out + row * D + nb * 16 + laneLo);
      }
    }
#endif
  }

  // ---- Tail (n % 16 atoms) — handled by block 0, simple path ------------
  const int rem = nTiles << 4;
  if (blockIdx.x == 0 && rem < n) {
    for (int a = rem + (int)(threadIdx.x >> 5); a < n;
         a += (int)(blockDim.x >> 5)) {
      int cb = ((int)threadIdx.x & 31) * 4;
      int k0 = i_atom[a], k1 = i_deg[a], k2 = i_chg[a];
      int k3 = i_hyb[a], k4 = i_nh[a], k5 = i_chi[a];
      float x0 = scalars[a * 3 + 0], x1 = scalars[a * 3 + 1],
            x2 = scalars[a * 3 + 2];
#pragma unroll
      for (int k = 0; k < 4; ++k) {
        int col = cb + k;
        float v = ldsT[OFF0 + k0 * D + col] + ldsT[OFF1 + k1 * D + col] +
                  ldsT[OFF2 + k2 * D + col] + ldsT[OFF3 + k3 * D + col] +
                  ldsT[OFF4 + k4 * D + col] + ldsT[OFF5 + k5 * D + col];
        v += x0 * W[col * 3 + 0] + x1 * W[col * 3 + 1] + x2 * W[col * 3 + 2] +
             bias[col];
        out[a * D + col] = v;
      }
    }
  }
}

extern "C" void kernel_launch(void* const* d_in, const int* in_sizes, int n_in,
                              void* d_out, int out_size, void* d_ws,
                              size_t ws_size, hipStream_t stream) {
  const int* i_atom = (const int*)d_in[0];
  const int* i_deg = (const int*)d_in[1];
  const int* i_chg = (const int*)d_in[2];
  const int* i_hyb = (const int*)d_in[3];
  const int* i_nh = (const int*)d_in[4];
  const int* i_chi = (const int*)d_in[5];
  const float* scalars = (const float*)d_in[6];
  const float* T0 = (const float*)d_in[7];
  const float* T1 = (const float*)d_in[8];
  const float* T2 = (const float*)d_in[9];
  const float* T3 = (const float*)d_in[10];
  const float* T4 = (const float*)d_in[11];
  const float* T5 = (const float*)d_in[12];
  const float* W = (const float*)d_in[13];
  const float* bias = (const float*)d_in[14];
  float* out = (float*)d_out;

  const int n = in_sizes[0];
  const int tiles = (n + 15) / 16;
  int blocks = (tiles + 7) / 8;  // 8 waves / block, 1+ tile each
  if (blocks > 1024) blocks = 1024;  // grid-stride; limit table-replication
  if (blocks < 1) blocks = 1;

  atom_feat_kernel<<<blocks, 256, 0, stream>>>(
      i_atom, i_deg, i_chg, i_hyb, i_nh, i_chi, scalars, T0, T1, T2, T3, T4,
      T5, W, bias, out, n);
}